// SageConvolution_68659347194330
// MI455X (gfx1250) — compile-verified
//
#include <hip/hip_runtime.h>

typedef __attribute__((ext_vector_type(2))) float v2f;
typedef __attribute__((ext_vector_type(4))) float v4f;
typedef __attribute__((ext_vector_type(8))) float v8f;

#define D     256   // D_IN == D_OUT
#define KC    8     // K-chunk staged in LDS per buffer
#define WROW  264   // LDS row pitch (floats): 2*264 % 64 == 16 banks -> the two
                    // half-wave K-pairs read disjoint bank ranges (no conflict)

// ---------------------------------------------------------------------------
// Kernel 1: fused dual GEMM, V_WMMA_F32_16X16X4_F32 (wave32).
//   h   = x @ W_l                      -> d_ws
//   out = x @ W_r + bias_l + bias_r   -> d_out
// Block = 256 threads = 8 waves = one 16-row tile x all 256 columns.
// Wave w owns column tiles n0=[32w,32w+16) and n1=[32w+16,32w+32); one A
// fragment feeds 4 WMMAs per K-step (2 col tiles x 2 weight matrices).
// Weight K-chunks (both matrices) are DMA'd into double-buffered LDS with
// GLOBAL_LOAD_ASYNC_TO_LDS_B128 (ASYNCcnt-paced), overlapping WMMA compute.
// ---------------------------------------------------------------------------
__global__ __launch_bounds__(256) void sage_gemm_kernel(
    const float* __restrict__ x,
    const float* __restrict__ Wl, const float* __restrict__ bl,
    const float* __restrict__ Wr, const float* __restrict__ br,
    float* __restrict__ h, float* __restrict__ out, int N)
{
  // [buffer][matrix][row*WROW + col] : 2*2*8*264*4 = 33,792 bytes
  __shared__ float ws[2][2][KC * WROW];

  const int tid  = threadIdx.x;
  const int wave = tid >> 5;
  const int lane = tid & 31;
  const int r0   = blockIdx.x * 16;
  if (r0 >= N) return;

  const int m  = lane & 15;            // A row / B-C-D column within tile
  const int kk = (lane >> 4) << 1;     // K-pair selector: 0 or 2
  const int hi = (lane >> 4);          // half-wave select for C/D rows

  const int n0 = (wave << 5) + m;      // wave's first column tile
  const int n1 = n0 + 16;              // wave's second column tile

  const float* xrow = x + (size_t)(r0 + m) * D;

  // ---- async DMA of one KC x 256 chunk of Wl and Wr into LDS buffer ------
  // 512 b128-slots per matrix / 256 threads -> 2 slots/thread/matrix.
  auto stage = [&](int kbase, int buf) {
#pragma unroll
    for (int s = 0; s < 2; ++s) {
      const int idx = tid + (s << 8);          // 0..511
      const int row = idx >> 6;                // 0..KC-1
      const int col = (idx & 63) << 2;         // 0..252 step 4
      // low 32 bits of a generic LDS address == DS byte offset
      const unsigned lofL =
          (unsigned)(size_t)(const void*)&ws[buf][0][row * WROW + col];
      const unsigned lofR =
          (unsigned)(size_t)(const void*)&ws[buf][1][row * WROW + col];
      const float* gL = Wl + (size_t)(kbase + row) * D + col;
      const float* gR = Wr + (size_t)(kbase + row) * D + col;
      asm volatile("global_load_async_to_lds_b128 %0, %1, off"
                   :: "v"(lofL), "v"(gL) : "memory");
      asm volatile("global_load_async_to_lds_b128 %0, %1, off"
                   :: "v"(lofR), "v"(gR) : "memory");
    }
  };

  v8f acc_h0 = {}, acc_o0 = {}, acc_h1 = {}, acc_o1 = {};

  stage(0, 0);                                  // prologue: chunk 0
  for (int c = 0; c < D / KC; ++c) {
    const int buf = c & 1;
    if (c + 1 < D / KC) {
      stage((c + 1) * KC, buf ^ 1);             // prefetch next chunk
      // 4 new issues outstanding; <=4 left  =>  chunk c complete (in-order)
      asm volatile("s_wait_asynccnt 0x4" ::: "memory");
    } else {
      asm volatile("s_wait_asynccnt 0x0" ::: "memory");
    }
    __syncthreads();                            // chunk c visible to all waves

    const int kbase = c * KC;
#pragma unroll
    for (int k4 = 0; k4 < KC; k4 += 4) {
      v2f a;                                    // shared A fragment (x)
      a.x = xrow[kbase + k4 + kk];
      a.y = xrow[kbase + k4 + kk + 1];
      const int rb = k4 + kk;
      v2f bL0, bR0, bL1, bR1;                   // B fragments from LDS
      bL0.x = ws[buf][0][rb * WROW + n0]; bL0.y = ws[buf][0][(rb + 1) * WROW + n0];
      bR0.x = ws[buf][1][rb * WROW + n0]; bR0.y = ws[buf][1][(rb + 1) * WROW + n0];
      bL1.x = ws[buf][0][rb * WROW + n1]; bL1.y = ws[buf][0][(rb + 1) * WROW + n1];
      bR1.x = ws[buf][1][rb * WROW + n1]; bR1.y = ws[buf][1][(rb + 1) * WROW + n1];
      acc_h0 = __builtin_amdgcn_wmma_f32_16x16x4_f32(
          false, a, false, bL0, (short)0, acc_h0, false, false);
      acc_o0 = __builtin_amdgcn_wmma_f32_16x16x4_f32(
          false, a, false, bR0, (short)0, acc_o0, false, false);
      acc_h1 = __builtin_amdgcn_wmma_f32_16x16x4_f32(
          false, a, false, bL1, (short)0, acc_h1, false, false);
      acc_o1 = __builtin_amdgcn_wmma_f32_16x16x4_f32(
          false, a, false, bR1, (short)0, acc_o1, false, false);
    }
    __syncthreads();  // all waves done reading buf before its next overwrite
  }

  // ---- epilogue: C/D layout -> global (VGPR v: M=v or v+8, N=lane&15) ----
  const float bias0 = bl[n0] + br[n0];
  const float bias1 = bl[n1] + br[n1];
  const int   mbase = r0 + (hi << 3);
#pragma unroll
  for (int v = 0; v < 8; ++v) {
    const size_t row = (size_t)(mbase + v) * D;
    h[row + n0]   = acc_h0[v];
    out[row + n0] = acc_o0[v] + bias0;
    h[row + n1]   = acc_h1[v];
    out[row + n1] = acc_o1[v] + bias1;
  }
}

// ---------------------------------------------------------------------------
// Kernel 2: edge aggregation  out[row] += val * h[col]  (h/out L2-resident:
// 102.4 MB each in 192 MB L2 -> L2-bandwidth bound, ~6.5 TB through L2).
// One wave per edge; lane i covers floats [4i,4i+4) and [128+4i,128+4i+4)
// -> two fully-coalesced 512B GLOBAL_LOAD_B128 sweeps, 8 non-returning
// global_atomic_add_f32 per lane, next gather row prefetched.
// ---------------------------------------------------------------------------
__global__ __launch_bounds__(256) void sage_edge_kernel(
    const int*   __restrict__ erow, const int* __restrict__ ecol,
    const float* __restrict__ eval, const float* __restrict__ h,
    float* __restrict__ out, int E)
{
  const int lane   = threadIdx.x & 31;
  const int wpb    = blockDim.x >> 5;
  const int stride = gridDim.x * wpb;
  int e = blockIdx.x * wpb + (threadIdx.x >> 5);

  for (; e < E; e += stride) {
    const int   col = ecol[e];
    const int   row = erow[e];
    const float val = eval[e];

    const int en = (e + stride < E) ? (e + stride) : e;
    __builtin_prefetch(h + (size_t)ecol[en] * D + lane * 8, 0, 0);

    const float* hr   = h   + (size_t)col * D;
    float*       orow = out + (size_t)row * D;

    v4f h0 = *(const v4f*)(hr + lane * 4);
    v4f h1 = *(const v4f*)(hr + 128 + lane * 4);
#pragma unroll
    for (int i = 0; i < 4; ++i) {
      unsafeAtomicAdd(orow + lane * 4 + i,       val * h0[i]);
      unsafeAtomicAdd(orow + 128 + lane * 4 + i, val * h1[i]);
    }
  }
}

// ---------------------------------------------------------------------------
extern "C" void kernel_launch(void* const* d_in, const int* in_sizes, int n_in,
                              void* d_out, int out_size, void* d_ws, size_t ws_size,
                              hipStream_t stream) {
  const float* x    = (const float*)d_in[0];
  const int*   erow = (const int*)  d_in[1];
  const int*   ecol = (const int*)  d_in[2];
  const float* eval = (const float*)d_in[3];
  const float* Wl   = (const float*)d_in[4];
  const float* bl   = (const float*)d_in[5];
  const float* Wr   = (const float*)d_in[6];
  const float* br   = (const float*)d_in[7];
  float* out = (float*)d_out;
  float* h   = (float*)d_ws;                 // N*256 floats = 102.4 MB scratch

  const int N = in_sizes[0] / D;             // 100000 (== 6250 * 16)
  const int E = in_sizes[1];                 // 3200000

  sage_gemm_kernel<<<(N + 15) / 16, 256, 0, stream>>>(x, Wl, bl, Wr, br, h, out, N);
  sage_edge_kernel<<<16384, 256, 0, stream>>>(erow, ecol, eval, h, out, E);
}